// Weight_Generation_58780922413710
// MI455X (gfx1250) — compile-verified
//
#include <hip/hip_runtime.h>

// CDNA5 / gfx1250, wave32. f32 WMMA path: V_WMMA_F32_16X16X4_F32.
typedef __attribute__((ext_vector_type(2))) float v2f;
typedef __attribute__((ext_vector_type(8))) float v8f;

#define WOFF 557568   // weight-output elements before y in d_out (242*2304)
#define NHW  1048576  // 1024*32*32

// ---------------------------------------------------------------------------
// GEMM1: h_in[242,1024] = z_all[242,64] @ w2[64,1024] + b2.
// One 16x16 tile per wave; 16 Mtiles x 64 Ntiles = 1024 waves.
// A-layout per ISA: lane L holds A[L%16, 4*kk + 2*(L/16) + v] for vgpr v.
// B-layout:         lane L holds B[4*kk + 2*(L/16) + v, L%16].
// C/D layout:       lane L vgpr r -> C[r + 8*(L/16), L%16].
// Out-of-range A rows are CLAMPED (not zeroed): a garbage A row only affects
// the same-index C row, which the store guard discards.
// ---------------------------------------------------------------------------
__global__ void gemm1_wmma(const float* __restrict__ A, const float* __restrict__ B,
                           const float* __restrict__ bias, float* __restrict__ C) {
  int wid  = (blockIdx.x * blockDim.x + threadIdx.x) >> 5;   // 0..1023 exact
  int lane = threadIdx.x & 31;
  int mt = wid >> 6, nt = wid & 63;
  int hi = lane >> 4, lm = lane & 15;
  int rowA  = mt * 16 + lm;
  int rowAc = rowA < 242 ? rowA : 241;          // clamp, avoid OOB reads
  int col   = nt * 16 + lm;

  const float* Arow = A + rowAc * 64 + 2 * hi;
  const float* Bcol = B + 2 * hi * 1024 + col;

  // Preload all fragments first: loads issue as clauses, single wait, then
  // 16 back-to-back accumulator-chained WMMAs.
  v2f a[16];
  float bx[16], by[16];
  #pragma unroll
  for (int kk = 0; kk < 16; ++kk)
    a[kk] = *(const v2f*)(Arow + 4 * kk);
  #pragma unroll
  for (int kk = 0; kk < 16; ++kk) {
    bx[kk] = Bcol[(4 * kk) * 1024];
    by[kk] = Bcol[(4 * kk + 1) * 1024];
  }

  v8f c = {};
  #pragma unroll
  for (int kk = 0; kk < 16; ++kk) {
    v2f b; b.x = bx[kk]; b.y = by[kk];
    c = __builtin_amdgcn_wmma_f32_16x16x4_f32(false, a[kk], false, b, (short)0, c, false, false);
  }

  float bv = bias[col];
  #pragma unroll
  for (int r = 0; r < 8; ++r) {
    int rowC = mt * 16 + r + 8 * hi;
    if (rowC < 242) C[rowC * 1024 + col] = c[r] + bv;
  }
}

// ---------------------------------------------------------------------------
// GEMM2 + fused weight assembly.
// A = h_in viewed as [3872,64]; B = w1 [64,144]; bias = b1 [144].
// h_fin row m = n*16 + o, col = i*9 + f  (kernels[n,o,i,fh,fw]).
// Tile mt covers rows n*16..n*16+15 with n == mt (wave-uniform layer decode).
// out[ offs(layer)*2304 + ((on*16+o)*(i_n*16) + in_*16 + i)*9 + f ].
// M = 3872 = 242*16 exact -> no row guards/clamps needed.
// ---------------------------------------------------------------------------
__global__ void gemm2_wmma_scatter(const float* __restrict__ A, const float* __restrict__ B,
                                   const float* __restrict__ bias, float* __restrict__ out) {
  int wid = (blockIdx.x * blockDim.x + threadIdx.x) >> 5;
  if (wid >= 2178) return;                      // wave-uniform exit
  int lane = threadIdx.x & 31;
  int mt = wid / 9, nt = wid - mt * 9;
  int hi = lane >> 4, lm = lane & 15;
  int rowA = mt * 16 + lm;
  int col  = nt * 16 + lm;

  const float* Arow = A + rowA * 64 + 2 * hi;
  const float* Bcol = B + 2 * hi * 144 + col;

  v2f a[16];
  float bx[16], by[16];
  #pragma unroll
  for (int kk = 0; kk < 16; ++kk)
    a[kk] = *(const v2f*)(Arow + 4 * kk);
  #pragma unroll
  for (int kk = 0; kk < 16; ++kk) {
    bx[kk] = Bcol[(4 * kk) * 144];
    by[kk] = Bcol[(4 * kk + 1) * 144];
  }

  v8f c = {};
  #pragma unroll
  for (int kk = 0; kk < 16; ++kk) {
    v2f b; b.x = bx[kk]; b.y = by[kk];
    c = __builtin_amdgcn_wmma_f32_16x16x4_f32(false, a[kk], false, b, (short)0, c, false, false);
  }
  float bv = bias[col];

  // layer decode for z-index n = mt (Z_NUM_LIST: 12x[1,1],[2,1],11x[2,2],[4,2],11x[4,4])
  int n = mt, local, i_n;
  if      (n < 12) { local = 0;              i_n = 1; }
  else if (n < 14) { local = n - 12;         i_n = 1; }
  else if (n < 58) { local = (n - 14) & 3;   i_n = 2; }
  else if (n < 66) { local = n - 58;         i_n = 2; }
  else             { local = (n - 66) & 15;  i_n = 4; }
  int nb  = n - local;                 // == _OFFS[layer]
  int on  = local / i_n;
  int in_ = local - on * i_n;
  int i = col / 9, f = col - i * 9;
  int w16x9 = i_n * 16 * 9;
  int base = nb * 2304 + on * 16 * w16x9 + (in_ * 16 + i) * 9 + f;
  #pragma unroll
  for (int r = 0; r < 8; ++r) {
    int o = r + 8 * hi;
    out[base + o * w16x9] = c[r] + bv;
  }
}

// ---------------------------------------------------------------------------
// Conv 3->16 3x3 SAME + per-channel sum/sumsq partials (deterministic tree).
// One thread per output pixel; pre-BN y written into its final d_out slot.
// ---------------------------------------------------------------------------
__global__ void conv_stats_kernel(const float* __restrict__ x, const float* __restrict__ cw,
                                  const float* __restrict__ cb, float* __restrict__ ypre,
                                  float* __restrict__ partials) {
  __shared__ float wl[432];
  __shared__ float bl[16];
  __shared__ float red[4096];
  __shared__ float red2[256];
  int tid = threadIdx.x;
  for (int k = tid; k < 432; k += 256) wl[k] = cw[k];
  if (tid < 16) bl[tid] = cb[tid];
  __syncthreads();

  int p = blockIdx.x * 256 + tid;              // < 1,048,576
  int n = p >> 10, hw = p & 1023, h = hw >> 5, w = hw & 31;
  const float* xb = x + n * 3072;
  float in[27];
  int idx = 0;
  #pragma unroll
  for (int ci = 0; ci < 3; ++ci)
    #pragma unroll
    for (int dh = -1; dh <= 1; ++dh)
      #pragma unroll
      for (int dw = -1; dw <= 1; ++dw) {
        int hh = h + dh, ww = w + dw;
        in[idx++] = ((unsigned)hh < 32u && (unsigned)ww < 32u) ? xb[ci * 1024 + hh * 32 + ww] : 0.f;
      }
  float acc[16];
  float* yo = ypre + n * 16384 + hw;
  #pragma unroll
  for (int oc = 0; oc < 16; ++oc) {
    float a = bl[oc];
    #pragma unroll
    for (int j = 0; j < 27; ++j) a = fmaf(in[j], wl[oc * 27 + j], a);
    acc[oc] = a;
    yo[oc * 1024] = a;                          // coalesced per-channel store
  }
  // deterministic block reduction: phase 0 = sum, phase 1 = sumsq
  for (int phase = 0; phase < 2; ++phase) {
    #pragma unroll
    for (int c0 = 0; c0 < 16; ++c0)
      red[c0 * 256 + tid] = phase ? acc[c0] * acc[c0] : acc[c0];
    __syncthreads();
    int c = tid & 15, chunk = tid >> 4;
    float pp = 0.f;
    #pragma unroll
    for (int j = 0; j < 16; ++j) pp += red[c * 256 + chunk * 16 + j];
    __syncthreads();
    red2[tid] = pp;                             // red2[chunk*16 + c]
    __syncthreads();
    if (tid < 16) {
      float t = 0.f;
      #pragma unroll
      for (int ch = 0; ch < 16; ++ch) t += red2[ch * 16 + tid];
      partials[blockIdx.x * 32 + phase * 16 + tid] = t;
    }
    __syncthreads();
  }
}

__global__ void reduce_partials(const float* __restrict__ partials, float* __restrict__ sums) {
  __shared__ float s[256];
  int j = blockIdx.x;                           // 0..31
  float loc = 0.f;
  for (int i = threadIdx.x; i < 4096; i += 256) loc += partials[i * 32 + j];
  s[threadIdx.x] = loc;
  __syncthreads();
  for (int st = 128; st > 0; st >>= 1) {
    if (threadIdx.x < st) s[threadIdx.x] += s[threadIdx.x + st];
    __syncthreads();
  }
  if (threadIdx.x == 0) sums[j] = s[0];
}

__global__ void finalize_stats(const float* __restrict__ sums, float* __restrict__ ss,
                               const float* __restrict__ gamma, const float* __restrict__ beta) {
  int c = threadIdx.x;
  if (c >= 16) return;
  const float invN = 1.f / (float)NHW;
  float mean = sums[c] * invN;
  float var  = sums[16 + c] * invN - mean * mean;   // population variance (jnp.var)
  float inv  = rsqrtf(var + 1e-5f);
  float sc   = gamma[c] * inv;
  ss[c]      = sc;
  ss[16 + c] = beta[c] - mean * sc;
}

__global__ void bnrelu_kernel(float* __restrict__ y, const float* __restrict__ ss) {
  int t = blockIdx.x * 256 + threadIdx.x;       // 4,194,304 float4s
  int oc = (t >> 8) & 15;                       // 256 float4s per (n,channel) plane
  float4* yp = reinterpret_cast<float4*>(y);
  float4 v = yp[t];
  float sc = ss[oc], sh = ss[16 + oc];
  v.x = fmaxf(fmaf(v.x, sc, sh), 0.f);
  v.y = fmaxf(fmaf(v.y, sc, sh), 0.f);
  v.z = fmaxf(fmaf(v.z, sc, sh), 0.f);
  v.w = fmaxf(fmaf(v.w, sc, sh), 0.f);
  yp[t] = v;
}

// ---------------------------------------------------------------------------
extern "C" void kernel_launch(void* const* d_in, const int* in_sizes, int n_in,
                              void* d_out, int out_size, void* d_ws, size_t ws_size,
                              hipStream_t stream) {
  const float* x      = (const float*)d_in[0];
  const float* conv_w = (const float*)d_in[1];
  const float* conv_b = (const float*)d_in[2];
  const float* gamma  = (const float*)d_in[3];
  const float* beta   = (const float*)d_in[4];
  const float* z_all  = (const float*)d_in[5];
  const float* w1     = (const float*)d_in[6];
  const float* b1     = (const float*)d_in[7];
  const float* w2     = (const float*)d_in[8];
  const float* b2     = (const float*)d_in[9];

  float* out = (float*)d_out;
  float* ws  = (float*)d_ws;
  float* h_in     = ws;            // 242*1024 = 247,808 floats
  float* partials = ws + 247808;   // 4096*32  = 131,072 floats
  float* sums     = ws + 378880;   // 32 floats
  float* ss       = ws + 378912;   // 32 floats (scale[16], shift[16])
  float* ypre     = out + WOFF;    // y region of d_out, used as pre-BN scratch

  // 1) h_in = z_all @ w2 + b2   (16 Mtiles x 64 Ntiles = 1024 waves)
  gemm1_wmma<<<128, 256, 0, stream>>>(z_all, w2, b2, h_in);
  // 2) h_fin = h_in @ w1 + b1, scattered straight into assembled-weight layout
  //    (242 Mtiles x 9 Ntiles = 2178 waves -> 273 blocks of 8 waves)
  gemm2_wmma_scatter<<<273, 256, 0, stream>>>(h_in, w1, b1, out);
  // 3) conv + per-block BN partial stats, pre-BN y into d_out
  conv_stats_kernel<<<4096, 256, 0, stream>>>(x, conv_w, conv_b, ypre, partials);
  // 4) reduce 4096 partials -> per-channel sum / sumsq
  reduce_partials<<<32, 256, 0, stream>>>(partials, sums);
  // 5) scale/shift from batch statistics
  finalize_stats<<<1, 16, 0, stream>>>(sums, ss, gamma, beta);
  // 6) in-place BN + ReLU over y (float4 vectorized)
  bnrelu_kernel<<<16384, 256, 0, stream>>>(ypre, ss);
}